// PerfectTreeTraversalTreeImpl_50302656970964
// MI455X (gfx1250) — compile-verified
//
#include <hip/hip_runtime.h>
#include <hip/hip_bf16.h>

// Problem constants (match reference)
#define B_      32768
#define T_      256
#define DEPTH_  10
#define F_      512
#define LVLPAD_ 131072          // T_ * 2^(DEPTH_-1)
#define NB_     8               // batch rows staged per workgroup (= ILP depth)
#define NTHREADS_ 256
#define NLDSLVL_ 3              // levels 1..3 tables staged in LDS
#define LDS_TAB_ 3584           // 512 + 1024 + 2048 entries

typedef unsigned int u32;
typedef u32 u32x4 __attribute__((ext_vector_type(4)));
typedef u32 u32x8 __attribute__((ext_vector_type(8)));

// ---------------------------------------------------------------------------
// CDNA5 async global -> LDS copy helpers (ASYNCcnt-tracked, ISA §10 / §15.18)
// ---------------------------------------------------------------------------
__device__ __forceinline__ unsigned lds_off_of(const void* p) {
  // Flat LDS addresses carry the LDS byte offset in the low 32 bits (ISA §10.2).
  return (unsigned)(unsigned long long)p;
}

__device__ __forceinline__ void async_b128(unsigned lds_off, const void* g) {
  asm volatile("global_load_async_to_lds_b128 %0, %1, off"
               :: "v"(lds_off), "v"((unsigned long long)g)
               : "memory");
}

// Cooperative region copy: 16B per lane-op, strided across the workgroup.
__device__ __forceinline__ void async_region(unsigned lds_base, const void* g, int bytes) {
  const char* gc = (const char*)g;
  for (int o = (int)threadIdx.x * 16; o < bytes; o += NTHREADS_ * 16)
    async_b128(lds_base + (unsigned)o, gc + o);
}

__device__ __forceinline__ void wait_async_all() {
  asm volatile("s_wait_asynccnt 0x0" ::: "memory");
}

// ---------------------------------------------------------------------------
// Tensor Data Mover: 1-D DMA of `ndw` dwords from global -> LDS (ISA §7/§8).
// Issue from ONE wave only (TDM ignores EXEC; guard with a whole-wave branch).
// ---------------------------------------------------------------------------
__device__ __forceinline__ void tdm_load_1d(unsigned lds_addr, const void* gptr, u32 ndw) {
  unsigned long long ga = (unsigned long long)gptr;
  u32x4 g0;
  g0[0] = 1u;                                             // count=1 (valid), flags=0
  g0[1] = lds_addr;                                       // lds_addr (bytes)
  g0[2] = (u32)(ga & 0xffffffffu);                        // global_addr[31:0]
  g0[3] = (u32)((ga >> 32) & 0x01ffffffu) | (2u << 30);   // global_addr[56:32], type=2

  u32x8 g1;
  g1[0] = (2u << 16);                                     // wg_mask=0, data_size=2 (4B)
  g1[1] = (ndw & 0xffffu) << 16;                          // tensor_dim0[15:0]
  g1[2] = ((ndw >> 16) & 0xffffu) | (1u << 16);           // tensor_dim0[31:16], tensor_dim1=1
  g1[3] = (ndw & 0xffffu) << 16;                          // tile_dim0 = ndw
  g1[4] = 1u;                                             // tile_dim1=1, tile_dim2=0
  g1[5] = ndw;                                            // tensor_dim0_stride[31:0]
  g1[6] = 0u;                                             // stride0[47:32], stride1[15:0]
  g1[7] = 0u;
  asm volatile("tensor_load_to_lds %0, %1" :: "s"(g0), "s"(g1) : "memory");
}

// ---------------------------------------------------------------------------
// Pre-pass: interleave (feature, bias) into 8B pairs in d_ws so each tree
// level costs one b64 load instead of two random loads.
// ---------------------------------------------------------------------------
__global__ __launch_bounds__(256) void pack_pairs_kernel(
    const int* __restrict__ nodes, const float* __restrict__ biases,
    int2* __restrict__ pw, int n)
{
  int i = blockIdx.x * blockDim.x + threadIdx.x;
  if (i < n) {
    int2 p;
    p.x = nodes[i];
    p.y = __float_as_int(biases[i]);
    pw[i] = p;
  }
}

// ---------------------------------------------------------------------------
// Main traversal kernel. threadIdx.x = tree id; each block owns NB_ batch rows.
// ---------------------------------------------------------------------------
template <bool PACKED>
__global__ __launch_bounds__(NTHREADS_) void tree_traverse_kernel(
    const float* __restrict__ x,
    const int*   __restrict__ root_nodes,
    const float* __restrict__ root_biases,
    const int*   __restrict__ nodes,
    const float* __restrict__ biases,
    const float* __restrict__ leaf,
    const int2*  __restrict__ pw,
    float*       __restrict__ out)
{
  __shared__ __attribute__((aligned(16))) float         xs[NB_ * F_];     // 16 KB
  __shared__ __attribute__((aligned(16))) unsigned char tabmem[LDS_TAB_ * 8]; // 28 KB
  __shared__ __attribute__((aligned(16))) int           rn[T_];           // 1 KB
  __shared__ __attribute__((aligned(16))) float         rb[T_];           // 1 KB

  const int t = (int)threadIdx.x;                 // tree index
  const size_t b0 = (size_t)blockIdx.x * NB_;     // first batch row

  // ---- TDM: x tile is one contiguous block (NB_ consecutive rows) ----
  if (threadIdx.x < 32) {                         // wave 0 only (whole-wave branch)
    tdm_load_1d(lds_off_of(xs), x + b0 * F_, NB_ * F_);
  }

  // ---- async stage: level 1-3 tables + root tables ----
  const unsigned tab = lds_off_of(tabmem);
  if (PACKED) {
    async_region(tab + 0,          pw + 0 * LVLPAD_,  512 * 8);
    async_region(tab + 512 * 8,    pw + 1 * LVLPAD_, 1024 * 8);
    async_region(tab + 1536 * 8,   pw + 2 * LVLPAD_, 2048 * 8);
  } else {
    async_region(tab + 0,               nodes  + 0 * LVLPAD_,  512 * 4);
    async_region(tab + 512 * 4,         nodes  + 1 * LVLPAD_, 1024 * 4);
    async_region(tab + 1536 * 4,        nodes  + 2 * LVLPAD_, 2048 * 4);
    async_region(tab + LDS_TAB_ * 4,             biases + 0 * LVLPAD_,  512 * 4);
    async_region(tab + LDS_TAB_ * 4 + 512 * 4,   biases + 1 * LVLPAD_, 1024 * 4);
    async_region(tab + LDS_TAB_ * 4 + 1536 * 4,  biases + 2 * LVLPAD_, 2048 * 4);
  }
  async_region(lds_off_of(rn), root_nodes,  T_ * 4);
  async_region(lds_off_of(rb), root_biases, T_ * 4);

  __builtin_amdgcn_s_wait_tensorcnt(0);
  wait_async_all();
  __syncthreads();

  const int2*  ltab = (const int2*)tabmem;                     // PACKED view
  const int*   ln   = (const int*)tabmem;                      // SOA view
  const float* lbp  = (const float*)(tabmem + LDS_TAB_ * 4);   // SOA view

  const int   rfeat = rn[t];
  const float rbias = rb[t];

  // All NB_ rows traversed concurrently: 8 independent loads per level.
  int prev[NB_];
#pragma unroll
  for (int k = 0; k < NB_; ++k) {
    float f = xs[k * F_ + rfeat];
    prev[k] = 2 * t + (f <= rbias ? 1 : 0);
  }

  // Levels 1..3: tables live in LDS
  int base = 0;
#pragma unroll
  for (int lvl = 1; lvl <= NLDSLVL_; ++lvl) {
#pragma unroll
    for (int k = 0; k < NB_; ++k) {
      int p = prev[k];
      int   feat;
      float bias;
      if (PACKED) {
        int2 nbp = ltab[base + p];            // one ds_load_b64
        feat = nbp.x;
        bias = __int_as_float(nbp.y);
      } else {
        feat = ln[base + p];
        bias = lbp[base + p];
      }
      float f = xs[k * F_ + feat];
      prev[k] = 2 * p + (f <= bias ? 1 : 0);
    }
    base += (T_ << lvl);
  }

  // Levels 4..9: L2-resident tables (packed pairs if ws was big enough)
#pragma unroll
  for (int lvl = NLDSLVL_ + 1; lvl < DEPTH_; ++lvl) {
    if (PACKED) {
      const int2* gtab = pw + (size_t)(lvl - 1) * LVLPAD_;
#pragma unroll
      for (int k = 0; k < NB_; ++k) {
        int  p  = prev[k];
        int2 nb = gtab[p];                    // one global_load_b64
        float f = xs[k * F_ + nb.x];
        prev[k] = 2 * p + (f <= __int_as_float(nb.y) ? 1 : 0);
      }
    } else {
      const int*   nt = nodes  + (size_t)(lvl - 1) * LVLPAD_;
      const float* bt = biases + (size_t)(lvl - 1) * LVLPAD_;
#pragma unroll
      for (int k = 0; k < NB_; ++k) {
        int   p    = prev[k];
        int   feat = nt[p];
        float bias = bt[p];
        float f    = xs[k * F_ + feat];
        prev[k] = 2 * p + (f <= bias ? 1 : 0);
      }
    }
  }

  // Leaf lookup (C == 1) + coalesced store (32 consecutive trees per wave)
#pragma unroll
  for (int k = 0; k < NB_; ++k)
    out[(b0 + (size_t)k) * T_ + t] = leaf[prev[k]];
}

// ---------------------------------------------------------------------------
extern "C" void kernel_launch(void* const* d_in, const int* in_sizes, int n_in,
                              void* d_out, int out_size, void* d_ws, size_t ws_size,
                              hipStream_t stream) {
  const float* x           = (const float*)d_in[0];
  const int*   root_nodes  = (const int*)  d_in[1];
  const float* root_biases = (const float*)d_in[2];
  const int*   nodes       = (const int*)  d_in[3];
  const float* biases      = (const float*)d_in[4];
  const float* leaf        = (const float*)d_in[5];
  float*       out         = (float*)d_out;

  const int    n_pairs  = (DEPTH_ - 1) * LVLPAD_;          // 1,179,648
  const size_t ws_need  = (size_t)n_pairs * sizeof(int2);  // ~9.4 MB
  const int    n_blocks = B_ / NB_;                        // 4096

  if (ws_size >= ws_need) {
    int2* pw = (int2*)d_ws;
    pack_pairs_kernel<<<(n_pairs + 255) / 256, 256, 0, stream>>>(nodes, biases, pw, n_pairs);
    tree_traverse_kernel<true><<<n_blocks, NTHREADS_, 0, stream>>>(
        x, root_nodes, root_biases, nodes, biases, leaf, pw, out);
  } else {
    tree_traverse_kernel<false><<<n_blocks, NTHREADS_, 0, stream>>>(
        x, root_nodes, root_biases, nodes, biases, leaf, (const int2*)nullptr, out);
  }
}